// HomoSAGEEncoder_88648124991294
// MI455X (gfx1250) — compile-verified
//
#include <hip/hip_runtime.h>
#include <hip/hip_bf16.h>

typedef __attribute__((ext_vector_type(2))) float v2f;
typedef __attribute__((ext_vector_type(8))) float v8f;

#define DIM 128
#define LDSPITCH 136   // 128 + 8 pad: removes LDS bank conflicts, keeps 16B align

// ---------------------------------------------------------------------------
// Degree: deg[v] += 1 for each edge (dst)
// ---------------------------------------------------------------------------
__launch_bounds__(256)
__global__ void sage_degree(const int* __restrict__ dst, float* __restrict__ deg, int E) {
    int i = blockIdx.x * 256 + threadIdx.x;
    if (i < E) atomicAdd(&deg[dst[i]], 1.0f);
}

// deg -> 1/max(deg,1)  (in place)
__launch_bounds__(256)
__global__ void sage_invdeg(float* __restrict__ deg, int N) {
    int i = blockIdx.x * 256 + threadIdx.x;
    if (i < N) {
        float d = deg[i];
        deg[i] = 1.0f / fmaxf(d, 1.0f);
    }
}

// ---------------------------------------------------------------------------
// Aggregate: one wave32 per edge. Lane l moves float4 (cols 4l..4l+3):
// coalesced 512B gather of h[src] (L2-resident), 4 fp32 atomics into acc[dst].
// ---------------------------------------------------------------------------
__launch_bounds__(256)
__global__ void sage_aggregate(const float* __restrict__ h,
                               const int* __restrict__ src,
                               const int* __restrict__ dst,
                               float* __restrict__ acc, int E) {
    int e = blockIdx.x * 8 + (threadIdx.x >> 5);
    if (e >= E) return;
    int lane = threadIdx.x & 31;
    int s = src[e];
    int d = dst[e];
    const float4 v = ((const float4*)(h + (size_t)s * DIM))[lane];
    float* a = acc + (size_t)d * DIM + lane * 4;
    atomicAdd(a + 0, v.x);
    atomicAdd(a + 1, v.y);
    atomicAdd(a + 2, v.z);
    atomicAdd(a + 3, v.w);
}

// ---------------------------------------------------------------------------
// Fused layer GEMM:
//   hout = ReLU( hin @ Wself + (acc * invdeg) @ Wneigh + bias )
// 256 threads = 8 waves; block stages Wself/Wneigh in LDS (padded), each wave
// computes a 16-row strip over all 128 output cols with fp32 WMMA 16x16x4.
//
// fp32 WMMA fragment layouts (wave32):
//   A 16x4 : lane = m + 16*(k>=2), vgpr v -> k = v + 2*(lane>>4)
//   B 4x16 : lane half selects k group likewise, n = lane&15
//   C 16x16: vgpr i -> m = i + 8*(lane>>4), n = lane&15
// ---------------------------------------------------------------------------
__launch_bounds__(256)
__global__ void sage_layer_gemm(const float* __restrict__ hin,
                                const float* __restrict__ acc,
                                const float* __restrict__ invdeg,
                                const float* __restrict__ Wself,
                                const float* __restrict__ Wneigh,
                                const float* __restrict__ bias,
                                float* __restrict__ hout, int N) {
    extern __shared__ float lds[];
    float* ldsS = lds;                    // [128][LDSPITCH]
    float* ldsN = lds + DIM * LDSPITCH;

    const int tid = threadIdx.x;

    // Cooperative weight staging: 128x128 floats each, float4 moves.
    for (int i = tid; i < DIM * (DIM / 4); i += 256) {
        int r  = i >> 5;          // row 0..127
        int c4 = (i & 31) << 2;   // col 0,4,...,124
        float4 ws = ((const float4*)Wself)[i];
        float4 wn = ((const float4*)Wneigh)[i];
        *(float4*)(ldsS + r * LDSPITCH + c4) = ws;
        *(float4*)(ldsN + r * LDSPITCH + c4) = wn;
    }
    __syncthreads();

    const int lane    = tid & 31;
    const int wave    = tid >> 5;
    const int rowBase = (blockIdx.x * 8 + wave) * 16;
    if (rowBase >= N) return;   // whole-wave uniform exit; EXEC stays all-1s

    const int n15   = lane & 15;
    const int khalf = (lane >> 4) << 1;            // 0 or 2
    int mrow = rowBase + n15;
    if (mrow >= N) mrow = N - 1;                   // safety clamp (N%16==0 normally)

    const float idg = invdeg[mrow];
    const float* hrow = hin + (size_t)mrow * DIM;
    const float* arow = acc + (size_t)mrow * DIM;

    // Preload all A fragments for both operands (held in registers across nt).
    v2f aS[32], aN[32];
#pragma unroll
    for (int kk = 0; kk < 32; ++kk) {
        int c0 = kk * 4 + khalf;                   // k = c0 + v, v in {0,1}
        v2f hs = *(const v2f*)(hrow + c0);
        v2f hn = *(const v2f*)(arow + c0);
        aS[kk] = hs;
        v2f sn;
        sn[0] = hn[0] * idg;
        sn[1] = hn[1] * idg;
        aN[kk] = sn;
    }

    for (int nt = 0; nt < 8; ++nt) {
        const float b = bias[nt * 16 + n15];
        v8f c;
#pragma unroll
        for (int i = 0; i < 8; ++i) c[i] = b;      // bias broadcast down rows

        const float* bsBase = ldsS + nt * 16 + n15;
        const float* bnBase = ldsN + nt * 16 + n15;
#pragma unroll
        for (int kk = 0; kk < 32; ++kk) {
            const int krow = kk * 4 + khalf;
            v2f bS, bN;
            bS[0] = bsBase[(krow + 0) * LDSPITCH];
            bS[1] = bsBase[(krow + 1) * LDSPITCH];
            bN[0] = bnBase[(krow + 0) * LDSPITCH];
            bN[1] = bnBase[(krow + 1) * LDSPITCH];
            c = __builtin_amdgcn_wmma_f32_16x16x4_f32(
                    false, aS[kk], false, bS, (short)0, c, false, false);
            c = __builtin_amdgcn_wmma_f32_16x16x4_f32(
                    false, aN[kk], false, bN, (short)0, c, false, false);
        }

        // ReLU + store D tile
#pragma unroll
        for (int i = 0; i < 8; ++i) {
            int m = rowBase + i + ((lane >> 4) << 3);
            float v = c[i];
            v = v > 0.0f ? v : 0.0f;
            if (m < N) hout[(size_t)m * DIM + nt * 16 + n15] = v;
        }
    }
}

// ---------------------------------------------------------------------------
// Driver
// ---------------------------------------------------------------------------
extern "C" void kernel_launch(void* const* d_in, const int* in_sizes, int n_in,
                              void* d_out, int out_size, void* d_ws, size_t ws_size,
                              hipStream_t stream) {
    const float* x      = (const float*)d_in[0];   // [N,128]
    const float* Wself  = (const float*)d_in[1];   // [L,128,128]
    const float* Wneigh = (const float*)d_in[2];   // [L,128,128]
    const float* bias   = (const float*)d_in[3];   // [L,128]
    const int*   src    = (const int*)d_in[4];     // [E]
    const int*   dst    = (const int*)d_in[5];     // [E]
    float*       out    = (float*)d_out;           // [N,128]

    const int N = in_sizes[0] / DIM;
    const int E = in_sizes[4];
    const int L = in_sizes[1] / (DIM * DIM);

    // Workspace layout: acc [N*128] | htmp [N*128] | deg/invdeg [N]
    float* acc  = (float*)d_ws;
    float* htmp = acc + (size_t)N * DIM;
    float* deg  = htmp + (size_t)N * DIM;

    // Degrees (shared across layers)
    hipMemsetAsync(deg, 0, (size_t)N * sizeof(float), stream);
    sage_degree<<<(E + 255) / 256, 256, 0, stream>>>(dst, deg, E);
    sage_invdeg<<<(N + 255) / 256, 256, 0, stream>>>(deg, N);

    const size_t ldsBytes = 2 * DIM * LDSPITCH * sizeof(float);
    const float* hin = x;
    for (int l = 0; l < L; ++l) {
        float* hout = (l == 1) ? htmp : out;   // x -> out -> htmp -> out
        hipMemsetAsync(acc, 0, (size_t)N * DIM * sizeof(float), stream);
        sage_aggregate<<<(E + 7) / 8, 256, 0, stream>>>(hin, src, dst, acc, E);
        sage_layer_gemm<<<(N + 127) / 128, 256, ldsBytes, stream>>>(
            hin, acc, deg,
            Wself  + (size_t)l * DIM * DIM,
            Wneigh + (size_t)l * DIM * DIM,
            bias   + (size_t)l * DIM,
            hout, N);
        hin = hout;
    }
}